// DnCNNablationFull_67697274520209
// MI455X (gfx1250) — compile-verified
//
#include <hip/hip_runtime.h>

typedef __attribute__((ext_vector_type(2))) float v2f;
typedef __attribute__((ext_vector_type(8))) float v8f;

#define TILE_H 16
#define TILE_W 32
#define HALO 2
#define XROWS (TILE_H + 2 * HALO)      // 20
#define XCOLS (TILE_W + 2 * HALO)      // 36
#define XSTRIDE 37                     // +1 pad vs 36 to avoid bank conflicts
#define XCH (XROWS * XSTRIDE)          // 740 floats per channel
#define KSTRIDE 513                    // 512 pixels + 1 pad
#define KROWS 26                       // 25 kernel channels + 1 trash row

// LDS offset of patch element for GEMM K-index kk = c*9 + ti*3 + tj
// (conv pad 1; x tile stored with halo 2, so tap (ti,tj) -> +(ti+1, tj+1)).
__host__ __device__ constexpr int koff(int kk) {
    return (kk / 9) * XCH + ((kk % 9) / 3 + 1) * XSTRIDE + ((kk % 9) % 3 + 1);
}

// Logical K-index permutation over physical WMMA K-slots (s = 2*ks + v):
//   lane half hi=0 uses kk = SLO[s];  hi=1 uses kk = SLO[s] + SDK[s],
// chosen so the LDS offset delta between halves is one of 3 constants:
//   group 0: +1 (tj+1), group 1: +XSTRIDE (ti+1), group 2: +XCH (c+1).
// Slot 9's hi half is the K=28 pad: A is zeroed there, B loads a don't-care.
__device__ __constant__ const int SLO[14] = {0,3,6,9,12,15,18,21,24,26, 2,11,20, 8};
#define SDK(s) ((s) == 9 ? 1000 : ((s) < 10 ? 1 : ((s) < 13 ? 3 : 9)))
#define SGRP(s) ((s) < 10 ? 0 : ((s) < 13 ? 1 : 2))
__host__ __device__ constexpr int slo(int s) {
    return (s==0)?0:(s==1)?3:(s==2)?6:(s==3)?9:(s==4)?12:(s==5)?15:(s==6)?18:
           (s==7)?21:(s==8)?24:(s==9)?26:(s==10)?2:(s==11)?11:(s==12)?20:8;
}

// One layer: feat_kernel 3x3 conv (3->25) via fp32 WMMA, PixelConv 5x5
// depthwise with per-pixel kernels, BatchNorm (running stats) + ReLU.
__global__ __launch_bounds__(256)
void pixconv_layer(const float* __restrict__ src, float* __restrict__ dst,
                   const float* __restrict__ W,  const float* __restrict__ bias,
                   const float* __restrict__ gamma, const float* __restrict__ beta,
                   const float* __restrict__ mean,  const float* __restrict__ var)
{
    __shared__ float xlds[3 * XCH];            // 2220 floats: x tile + halo 2
    __shared__ float klds[KROWS * KSTRIDE];    // per-pixel kernels + trash row 25
    __shared__ float wlds[675];                // W' [25][27]
    __shared__ float blds[25];

    const int tid  = threadIdx.x;
    const int img  = blockIdx.z;
    const int row0 = blockIdx.y * TILE_H;
    const int col0 = blockIdx.x * TILE_W;

    // ---- Phase 0: stage x tile (zero-padded halo) + layer weights into LDS ----
    for (int idx = tid; idx < 3 * XROWS * XCOLS; idx += 256) {
        int c  = idx / (XROWS * XCOLS);
        int r  = idx % (XROWS * XCOLS);
        int rr = r / XCOLS, cc = r % XCOLS;
        int gh = row0 + rr - HALO, gw = col0 + cc - HALO;
        float v = 0.f;
        if (gh >= 0 && gh < 256 && gw >= 0 && gw < 256)
            v = src[((img * 3 + c) * 256 + gh) * 256 + gw];
        xlds[c * XCH + rr * XSTRIDE + cc] = v;
    }
    for (int idx = tid; idx < 700; idx += 256) {
        if (idx < 675) wlds[idx] = W[idx];
        else           blds[idx - 675] = bias[idx - 675];
    }
    __syncthreads();

    // ---- Phase 1: k[25 x 512] = W'[25x27(pad28)] * patches[27x512] via WMMA ----
    const int lane = tid & 31;
    const int wave = tid >> 5;
    const int n    = lane & 15;   // M row / N column within half-wave
    const int hi   = lane >> 4;   // lane half selects K pair (f32 16x16x4 layout)

    // A fragments under the slot permutation (one-time setup)
    float afrag[2][7][2];
#pragma unroll
    for (int mt = 0; mt < 2; ++mt)
#pragma unroll
        for (int ks = 0; ks < 7; ++ks)
#pragma unroll
            for (int v = 0; v < 2; ++v) {
                int s  = 2 * ks + v;
                int p  = n + 16 * mt;
                int kk = hi ? (SLO[s] + SDK(s)) : SLO[s];
                afrag[mt][ks][v] = (p < 25 && kk < 27) ? wlds[p * 27 + kk] : 0.f;
            }

    // klds store rows: c0 -> r + 8*hi ; c1 -> r+16 (hi=0) / r==0?24:trash (hi=1)
    int srow1[8];
#pragma unroll
    for (int r = 0; r < 8; ++r)
        srow1[r] = (hi ? (r == 0 ? 24 : 25) : (r + 16)) * KSTRIDE;

    // Bias-initialized accumulators (identical for every N-tile)
    v8f cinit0, cinit1;
#pragma unroll
    for (int r = 0; r < 8; ++r) {
        int p = r + 8 * hi;                   // D layout: M = r + 8*(lane>=16)
        cinit0[r] = blds[p];
        cinit1[r] = (p + 16 < 25) ? blds[p + 16] : 0.f;
    }

#pragma unroll
    for (int t = 0; t < 4; ++t) {             // 8 waves * 4 = 32 N-tiles
        int q   = (wave * 4 + t) * 16 + n;    // pixel index 0..511
        int row = q >> 5, col = q & 31;
        const float* xb  = &xlds[row * XSTRIDE + col];
        // Three hi-shifted bases: all 14 B-gathers use immediate offsets.
        const float* xbg[3] = { xb + hi * 1, xb + hi * XSTRIDE, xb + hi * XCH };

        v8f c0 = cinit0, c1 = cinit1;
#pragma unroll
        for (int ks = 0; ks < 7; ++ks) {
            v2f b;
            b.x = xbg[SGRP(2 * ks + 0)][koff(slo(2 * ks + 0))];
            b.y = xbg[SGRP(2 * ks + 1)][koff(slo(2 * ks + 1))];
            v2f a0; a0.x = afrag[0][ks][0]; a0.y = afrag[0][ks][1];
            v2f a1; a1.x = afrag[1][ks][0]; a1.y = afrag[1][ks][1];
            c0 = __builtin_amdgcn_wmma_f32_16x16x4_f32(false, a0, false, b,
                                                       (short)0, c0, false, false);
            c1 = __builtin_amdgcn_wmma_f32_16x16x4_f32(false, a1, false, b,
                                                       (short)0, c1, false, false);
        }
        // Unconditional stores; invalid c1 rows (which are exactly 0) go to
        // trash row 25. No exec divergence between tiles.
        float* kb0 = &klds[8 * hi * KSTRIDE + q];
        float* kb1 = &klds[q];
#pragma unroll
        for (int r = 0; r < 8; ++r) {
            kb0[r * KSTRIDE] = c0[r];
            kb1[srow1[r]]    = c1[r];
        }
    }
    __syncthreads();

    // ---- Phase 2: PixelConv (per-pixel 5x5 depthwise) + BN + ReLU ----
    // Each thread: 2 adjacent pixels, all 3 channels. The per-pixel kernel
    // k[25] is channel-shared; adjacent pixels share 5x4 of the 5x5 x-patch.
    float sc[3], sh[3];
#pragma unroll
    for (int c = 0; c < 3; ++c) {
        sc[c] = gamma[c] * rsqrtf(var[c] + 1e-5f);
        sh[c] = beta[c] - mean[c] * sc[c];
    }
    {
        int q0  = 2 * tid;                    // even; q0,q0+1 same tile row
        int row = q0 >> 5, col = q0 & 31;
        const float* kb = &klds[q0];
        const float* xb = &xlds[row * XSTRIDE + col];
        float y[2][3] = {};
#pragma unroll
        for (int i = 0; i < 5; ++i) {
            float xv[3][6];
#pragma unroll
            for (int c = 0; c < 3; ++c)
#pragma unroll
                for (int j = 0; j < 6; ++j)
                    xv[c][j] = xb[c * XCH + i * XSTRIDE + j];
#pragma unroll
            for (int j = 0; j < 5; ++j) {
                int p = i * 5 + j;
                float k0 = kb[p * KSTRIDE];
                float k1 = kb[p * KSTRIDE + 1];
#pragma unroll
                for (int c = 0; c < 3; ++c) {
                    y[0][c] += xv[c][j]     * k0;
                    y[1][c] += xv[c][j + 1] * k1;
                }
            }
        }
        int gidx = ((img * 3) * 256 + (row0 + row)) * 256 + (col0 + col);
#pragma unroll
        for (int c = 0; c < 3; ++c) {
            dst[gidx + c * 65536]     = fmaxf(y[0][c] * sc[c] + sh[c], 0.f);
            dst[gidx + c * 65536 + 1] = fmaxf(y[1][c] * sc[c] + sh[c], 0.f);
        }
    }
}

extern "C" void kernel_launch(void* const* d_in, const int* in_sizes, int n_in,
                              void* d_out, int out_size, void* d_ws, size_t ws_size,
                              hipStream_t stream) {
    const float* x     = (const float*)d_in[0];  // [8,3,256,256]
    const float* Wk    = (const float*)d_in[1];  // [14,25,3,3,3]
    const float* bk    = (const float*)d_in[2];  // [14,25]
    const float* gam   = (const float*)d_in[3];  // [14,3]
    const float* bet   = (const float*)d_in[4];  // [14,3]
    const float* rmean = (const float*)d_in[5];  // [14,3]
    const float* rvar  = (const float*)d_in[6];  // [14,3]
    float* out = (float*)d_out;
    float* ws  = (float*)d_ws;                   // one 6.3 MB ping buffer

    dim3 grid(256 / TILE_W, 256 / TILE_H, 8);    // 8 x 16 x 8 = 1024 blocks
    dim3 block(256);

    // 14 layers, one launch each (grid-wide dependency between layers).
    // Even layers write ws, odd layers write d_out -> layer 13 lands in d_out.
    // Both buffers (12.6 MB) stay resident in the 192 MB L2 across layers.
    for (int l = 0; l < 14; ++l) {
        const float* srcp = (l == 0) ? x : ((l & 1) ? ws : out);
        float*       dstp = (l & 1) ? out : ws;
        pixconv_layer<<<grid, block, 0, stream>>>(
            srcp, dstp, Wk + l * 675, bk + l * 25,
            gam + l * 3, bet + l * 3, rmean + l * 3, rvar + l * 3);
    }
}